// Mel_Decoder_78400333021565
// MI455X (gfx1250) — compile-verified
//
#include <hip/hip_runtime.h>

// ---------------------------------------------------------------------------
// Mel decoder scan for MI455X (gfx1250, wave32).
// Scan kernel grid: 128 WGs = 8 batch-groups x 16 "slice" WGs.
// Each group owns 16 batch elements; slices cooperate through L2 buffers with
// sense-reversing atomic barriers (4 per step). All per-step GEMMs use
// v_wmma_f32_16x16x4_f32 with M=16 batch rows. State reloads use
// global_load_async_to_lds_b128 (ASYNCcnt) on gfx1250.
// ---------------------------------------------------------------------------

#define HD      256          // hidden
#define NSLICE  16           // WGs per group
#define NGROUP  8            // batch groups (8 * 16 = 128 batch)
#define TENC    512
#define TCH     32           // TENC / NSLICE
#define STEPS   200
#define BSZ     128
#define MELD    80

// per-group workspace layout (in floats)
enum : int { D0_ = 0, H1_ = 8192, H2_ = 16384, IN2_ = 24576, IO2_ = 32768,
             CP_ = 40960 /*16*16*256*/, SP_ = 106496 /*256*/, GSZ_ = 106752 };

typedef float v2f __attribute__((ext_vector_type(2)));
typedef float v8f __attribute__((ext_vector_type(8)));
typedef int   v4i __attribute__((ext_vector_type(4)));

struct Params {
    const float *enc, *xsrc_unused;
    const float *w2, *b2, *v_w, *v_b, *proj_w, *proj_b, *out_w, *out_b;
    const float *att_wih, *att_whh, *att_bih, *att_bhh;
    const float *g1_wih, *g1_whh, *g1_bih, *g1_bhh;
    const float *g2_wih, *g2_whh, *g2_bih, *g2_bhh;
    float *xs;      // [200][128][128] prenet outputs (precomputed)
    float *w1enc;   // [128][512][256] (precomputed)
    float *gws;     // per-group comm buffers
    unsigned *bar;  // barrier objects
    float *out;     // [128][1000][80]
};

// ---------------- math helpers (CDNA5 fast paths) --------------------------
__device__ __forceinline__ float fast_exp(float x) {
    return __builtin_amdgcn_exp2f(x * 1.44269504088896f);
}
__device__ __forceinline__ float fast_sig(float x) {
    return __builtin_amdgcn_rcpf(1.0f + fast_exp(-x));
}
#if __has_builtin(__builtin_amdgcn_tanhf)
__device__ __forceinline__ float fast_tanh(float x) { return __builtin_amdgcn_tanhf(x); }
#else
__device__ __forceinline__ float fast_tanh(float x) {
    float e = fast_exp(2.0f * x);
    return 1.0f - 2.0f * __builtin_amdgcn_rcpf(e + 1.0f);
}
#endif

// ---------------- async LDS copy (CDNA5 GLOBAL_LOAD_ASYNC_TO_LDS) ----------
#if __has_builtin(__builtin_amdgcn_global_load_async_to_lds_b128) && \
    __has_builtin(__builtin_amdgcn_s_wait_asynccnt)
#define HAVE_ASYNC_LDS 1
#endif
#ifndef HAVE_ASYNC_LDS
#define HAVE_ASYNC_LDS 0
#endif

// issue a global->LDS copy of nfloats (multiple of 4; 256 threads assumed)
__device__ __forceinline__ void lds_copy_issue(float* dst, const float* src, int nfloats) {
    for (int i = threadIdx.x * 4; i < nfloats; i += 1024) {
#if HAVE_ASYNC_LDS
        __builtin_amdgcn_global_load_async_to_lds_b128(
            (v4i*)(src + i), (v4i*)(dst + i), 0, 0);
#else
        *(float4*)(dst + i) = *(const float4*)(src + i);
#endif
    }
}
__device__ __forceinline__ void lds_copy_join() {
#if HAVE_ASYNC_LDS
    __builtin_amdgcn_s_wait_asynccnt(0);
#endif
}

// ---------------- WMMA tile helpers ----------------------------------------
// D[16,16] += A[16,K] * W[n0..n0+16, kofs..kofs+K]^T using v_wmma_f32_16x16x4.
// A layout (16x4):  lane<16 -> K{0,1}, lane>=16 -> K{2,3}; row = lane&15.
// B layout (4x16):  same K split; col n = lane&15. W is row-major [out, in],
// so B element (k,n) = W[n][k]: a float2 load per lane.
__device__ __forceinline__ v8f wmma_acc(const float* A, int lda,
                                        const float* W, int ldw,
                                        int n0, int kofs, int K, v8f acc) {
    const int lane = threadIdx.x & 31;
    const int ml   = lane & 15;
    const int ks   = (lane >> 4) << 1;
    const float* ap = A + ml * lda + ks;
    const float* wp = W + (size_t)(n0 + ml) * ldw + kofs + ks;
    v8f acc2 = {0.f, 0.f, 0.f, 0.f, 0.f, 0.f, 0.f, 0.f};
    const int Kh = K >> 1;                 // two accumulators hide WMMA RAW latency
#pragma unroll 4
    for (int k = 0; k < Kh; k += 4) {
        v2f a0 = *(const v2f*)(ap + k);
        v2f b0 = *(const v2f*)(wp + k);
        acc  = __builtin_amdgcn_wmma_f32_16x16x4_f32(false, a0, false, b0, (short)0, acc,  false, false);
        v2f a1 = *(const v2f*)(ap + Kh + k);
        v2f b1 = *(const v2f*)(wp + Kh + k);
        acc2 = __builtin_amdgcn_wmma_f32_16x16x4_f32(false, a1, false, b1, (short)0, acc2, false, false);
    }
    return acc + acc2;
}

// D layout: VGPR r holds row (r + 8*(lane>=16)), col = lane&15
__device__ __forceinline__ void store_tile(float* D, int ldd, int n0, v8f acc) {
    const int lane = threadIdx.x & 31;
    const int nl = lane & 15;
    const int r8 = (lane >> 4) * 8;
#pragma unroll
    for (int r = 0; r < 8; ++r) D[(r8 + r) * ldd + n0 + nl] = acc[r];
}

// ---------------- inter-WG barrier (agent scope, sense reversing) ----------
__device__ __forceinline__ void group_barrier(unsigned* cnt, unsigned* gen) {
    __threadfence();                 // publish this WG's global stores
    __syncthreads();
    if (threadIdx.x == 0) {
        unsigned g = __hip_atomic_load(gen, __ATOMIC_RELAXED, __HIP_MEMORY_SCOPE_AGENT);
        unsigned a = __hip_atomic_fetch_add(cnt, 1u, __ATOMIC_ACQ_REL, __HIP_MEMORY_SCOPE_AGENT);
        if (a == NSLICE - 1) {
            __hip_atomic_store(cnt, 0u, __ATOMIC_RELAXED, __HIP_MEMORY_SCOPE_AGENT);
            __hip_atomic_fetch_add(gen, 1u, __ATOMIC_RELEASE, __HIP_MEMORY_SCOPE_AGENT);
        } else {
            while (__hip_atomic_load(gen, __ATOMIC_ACQUIRE, __HIP_MEMORY_SCOPE_AGENT) == g)
                __builtin_amdgcn_s_sleep(1);
        }
    }
    __syncthreads();
    // per-wave acquire so every CU's near cache is invalidated before reloads
    if ((threadIdx.x & 31) == 0)
        (void)__hip_atomic_load(gen, __ATOMIC_ACQUIRE, __HIP_MEMORY_SCOPE_AGENT);
    __syncthreads();
}

// GRU gate tiles for slice j: waves 0..3 compute (r, z, i_n, h_n) 16x16 tiles.
__device__ __forceinline__ void gru_gates(const float* xa, int Ki, const float* ha,
                                          const float* Wi, const float* Wh,
                                          const float* bih, const float* bhh,
                                          float* g4, int j) {
    const int wv = threadIdx.x >> 5;
    if (wv < 4) {
        const int base = (wv == 0 ? 0 : (wv == 1 ? 256 : 512)) + j * 16;
        const int nl = threadIdx.x & 15;
        float bv;
        if (wv <= 1)      bv = bih[base + nl] + bhh[base + nl];  // r, z: ir+hr fused
        else if (wv == 2) bv = bih[base + nl];                    // i_n
        else              bv = bhh[base + nl];                    // h_n
        v8f acc;
#pragma unroll
        for (int r = 0; r < 8; ++r) acc[r] = bv;
        if (wv < 3)  acc = wmma_acc(xa, Ki, Wi, Ki, base, 0, Ki, acc);   // x @ Wi^T
        if (wv != 2) acc = wmma_acc(ha, HD, Wh, HD, base, 0, HD, acc);   // h @ Wh^T
        store_tile(g4 + wv * 256, 16, 0, acc);
    }
}

// ---------------------------------------------------------------------------
__global__ __launch_bounds__(256, 1) void scan_kernel(Params P) {
    __shared__ struct __align__(16) {
        float d[16][HD];
        float h1[16][HD];
        float h2[16][HD];
        float in2[16][HD];
        union { float w2d[16][HD];  float ddot[16][HD]; };  // disjoint lifetimes
        union { float p[16][HD];    float io2[16][HD];  };  // disjoint lifetimes
        float x[16][128];
        float g4[4][16][16];
        float esc[16][TCH];
        float vv[HD];
        float sigma[16];
    } S;

    const int tid  = threadIdx.x;
    const int wv   = tid >> 5;
    const int lane = tid & 31;
    const int g    = blockIdx.x / NSLICE;
    const int j    = blockIdx.x % NSLICE;
    const int gb   = g * 16;

    float*    gw  = P.gws + (size_t)g * GSZ_;
    unsigned* cnt = P.bar + g * 32;
    unsigned* gen = cnt + 1;

    for (int i = tid; i < 16 * HD; i += 256) {
        ((float*)S.d)[i] = 0.f; ((float*)S.h1)[i] = 0.f; ((float*)S.h2)[i] = 0.f;
    }
    S.vv[tid] = P.v_w[tid];
    const float vb = P.v_b[0];
    __syncthreads();

    for (int step = 0; step < STEPS; ++step) {
        const int par = step & 1;

        // stage x_i (prenet output) [16][128]
        lds_copy_issue((float*)S.x, P.xs + ((size_t)step * BSZ + gb) * 128, 16 * 128);
        lds_copy_join();
        __syncthreads();

        // ---- Phase A: attention-GRU, slice j of new d -----------------------
        gru_gates((const float*)S.x, 128, (const float*)S.d,
                  P.att_wih, P.att_whh, P.att_bih, P.att_bhh, &S.g4[0][0][0], j);
        __syncthreads();
        {
            int b = tid >> 4, c = tid & 15, col = j * 16 + c;
            float r = fast_sig(S.g4[0][b][c]);
            float z = fast_sig(S.g4[1][b][c]);
            float n = fast_tanh(S.g4[2][b][c] + r * S.g4[3][b][c]);
            gw[D0_ + par * 4096 + b * HD + col] = (1.f - z) * n + z * S.d[b][col];
        }
        group_barrier(cnt, gen);                       // bar1: d complete
        lds_copy_issue((float*)S.d, gw + D0_ + par * 4096, 4096);
        lds_copy_join();
        __syncthreads();

        // ---- w2d = d @ w2^T + b2 (redundant per WG; 16 tiles over 8 waves) --
        for (int tI = wv; tI < 16; tI += 8) {
            int n0 = tI * 16;
            v8f acc; float bvv = P.b2[n0 + (tid & 15)];
#pragma unroll
            for (int r = 0; r < 8; ++r) acc[r] = bvv;
            acc = wmma_acc((const float*)S.d, HD, P.w2, HD, n0, 0, HD, acc);
            store_tile((float*)S.w2d, HD, n0, acc);
        }
        __syncthreads();

        // ---- Phase B: attention scores for this WG's T-chunk ---------------
        for (int rr = 0; rr < 64; ++rr) {
            int row = rr * 8 + wv;                     // 512 (b,t) rows / 8 waves
            int b = row >> 5, tl = row & 31;
            const float* wrow = P.w1enc + (((size_t)(gb + b)) * TENC + (j * TCH + tl)) * HD + lane * 8;
            if (rr + 1 < 64) __builtin_prefetch(wrow + 8 * HD, 0, 1);
            const float* wd = &S.w2d[b][lane * 8];
            const float* vp = &S.vv[lane * 8];
            float s = 0.f;
#pragma unroll
            for (int q = 0; q < 8; ++q) s += vp[q] * fast_tanh(wrow[q] + wd[q]);
#pragma unroll
            for (int m = 16; m >= 1; m >>= 1) s += __shfl_xor(s, m, 32);
            if (lane == 0) S.esc[b][tl] = fast_exp(s + vb);   // bounded scores: no max pass
        }
        __syncthreads();
        if (tid < 16) {                                 // partial softmax denominator
            float s = 0.f;
            for (int tl = 0; tl < TCH; ++tl) s += S.esc[tid][tl];
            gw[SP_ + j * 16 + tid] = s;
        }
        {                                               // partial context
            int b = tid >> 4, h0 = (tid & 15) * 16;
            float acc[16];
#pragma unroll
            for (int q = 0; q < 16; ++q) acc[q] = 0.f;
            for (int tl = 0; tl < TCH; ++tl) {
                float e = S.esc[b][tl];
                const float* er = P.enc + (((size_t)(gb + b)) * TENC + (j * TCH + tl)) * HD + h0;
#pragma unroll
                for (int q = 0; q < 16; ++q) acc[q] += e * er[q];
            }
            float* cp = gw + CP_ + ((size_t)(j * 16 + b)) * HD + h0;
#pragma unroll
            for (int q = 0; q < 16; ++q) cp[q] = acc[q];
        }
        group_barrier(cnt, gen);                       // bar2: attention partials

        // ---- Phase C: ddot (redundant), p (redundant), GRU1 slice ----------
        if (tid < 16) {
            float s = 0.f;
            for (int jj = 0; jj < NSLICE; ++jj) s += gw[SP_ + jj * 16 + tid];
            S.sigma[tid] = s;
        }
        __syncthreads();
        {
            int b = tid >> 4, h0 = (tid & 15) * 16;
            float inv = __builtin_amdgcn_rcpf(S.sigma[b]);
            float acc[16];
#pragma unroll
            for (int q = 0; q < 16; ++q) acc[q] = 0.f;
            for (int jj = 0; jj < NSLICE; ++jj) {
                const float* cp = gw + CP_ + ((size_t)(jj * 16 + b)) * HD + h0;
#pragma unroll
                for (int q = 0; q < 16; ++q) acc[q] += cp[q];
            }
#pragma unroll
            for (int q = 0; q < 16; ++q) S.ddot[b][h0 + q] = acc[q] * inv;
        }
        __syncthreads();
        for (int tI = wv; tI < 16; tI += 8) {          // p = [d|ddot] @ proj^T + b
            int n0 = tI * 16;
            v8f acc; float bvv = P.proj_b[n0 + (tid & 15)];
#pragma unroll
            for (int r = 0; r < 8; ++r) acc[r] = bvv;
            acc = wmma_acc((const float*)S.d,    HD, P.proj_w, 512, n0, 0,   HD, acc);
            acc = wmma_acc((const float*)S.ddot, HD, P.proj_w, 512, n0, 256, HD, acc);
            store_tile((float*)S.p, HD, n0, acc);
        }
        __syncthreads();
        gru_gates((const float*)S.p, HD, (const float*)S.h1,
                  P.g1_wih, P.g1_whh, P.g1_bih, P.g1_bhh, &S.g4[0][0][0], j);
        __syncthreads();
        {
            int b = tid >> 4, c = tid & 15, col = j * 16 + c;
            float r = fast_sig(S.g4[0][b][c]);
            float z = fast_sig(S.g4[1][b][c]);
            float n = fast_tanh(S.g4[2][b][c] + r * S.g4[3][b][c]);
            float o1 = (1.f - z) * n + z * S.h1[b][col];
            gw[H1_  + par * 4096 + b * HD + col] = o1;
            gw[IN2_ + par * 4096 + b * HD + col] = o1 + S.p[b][col];
        }
        group_barrier(cnt, gen);                       // bar3: h1 / in2 complete
        lds_copy_issue((float*)S.h1,  gw + H1_  + par * 4096, 4096);
        lds_copy_issue((float*)S.in2, gw + IN2_ + par * 4096, 4096);
        lds_copy_join();
        __syncthreads();

        // ---- Phase D: GRU2 slice -------------------------------------------
        gru_gates((const float*)S.in2, HD, (const float*)S.h2,
                  P.g2_wih, P.g2_whh, P.g2_bih, P.g2_bhh, &S.g4[0][0][0], j);
        __syncthreads();
        {
            int b = tid >> 4, c = tid & 15, col = j * 16 + c;
            float r = fast_sig(S.g4[0][b][c]);
            float z = fast_sig(S.g4[1][b][c]);
            float n = fast_tanh(S.g4[2][b][c] + r * S.g4[3][b][c]);
            float o2 = (1.f - z) * n + z * S.h2[b][col];
            gw[H2_  + par * 4096 + b * HD + col] = o2;
            gw[IO2_ + par * 4096 + b * HD + col] = o2 + S.in2[b][col];
        }
        group_barrier(cnt, gen);                       // bar4: h2 / (in2+o2) complete
        lds_copy_issue((float*)S.h2,  gw + H2_  + par * 4096, 4096);
        lds_copy_issue((float*)S.io2, gw + IO2_ + par * 4096, 4096);
        lds_copy_join();
        __syncthreads();

        // ---- Phase E: out = (in2+o2) @ out_w^T + out_b (25 tiles = 400) ----
        {
            int slot = j * 8 + wv;
            if (slot < 25) {
                int n0 = slot * 16;
                v8f acc; float bvv = P.out_b[n0 + (tid & 15)];
#pragma unroll
                for (int r = 0; r < 8; ++r) acc[r] = bvv;
                acc = wmma_acc((const float*)S.io2, HD, P.out_w, HD, n0, 0, HD, acc);
                const int nl = lane & 15, r8 = (lane >> 4) * 8;
#pragma unroll
                for (int r = 0; r < 8; ++r) {
                    int b = r8 + r;
                    P.out[(size_t)(gb + b) * 80000 + (size_t)step * 400 + n0 + nl] = acc[r];
                }
            }
        }
        // no trailing barrier needed: double-buffered slots + bar1..bar4 spacing
    }
}

// ---------------------------------------------------------------------------
// Precompute kernels
// ---------------------------------------------------------------------------
__global__ __launch_bounds__(256) void prenet_kernel(const float* dec,
        const float* W1, const float* B1, const float* W2, const float* B2, float* xs) {
    __shared__ float inrow[8][80];
    __shared__ float p1[8][264];
    const int wvi = threadIdx.x >> 5, lane = threadIdx.x & 31;
    const int pair = blockIdx.x * 8 + wvi;       // 25600 = 200 steps * 128 batch
    const int s = pair >> 7, b = pair & 127;
    const float* drow = dec + ((size_t)b * 1000 + s * 5) * MELD;
    for (int k = lane; k < 80; k += 32) inrow[wvi][k] = drow[k];
    __syncthreads();
#pragma unroll
    for (int q = 0; q < 8; ++q) {
        int n = q * 32 + lane;
        float acc = B1[n];
        const float* wr = W1 + n * 80;
        for (int k = 0; k < 80; ++k) acc += inrow[wvi][k] * wr[k];
        p1[wvi][n] = fmaxf(acc, 0.f);
    }
    __syncthreads();
#pragma unroll
    for (int q = 0; q < 4; ++q) {
        int n = q * 32 + lane;
        float acc = B2[n];
        const float* wr = W2 + n * 256;
        for (int k = 0; k < 256; ++k) acc += p1[wvi][k] * wr[k];
        xs[((size_t)s * BSZ + b) * 128 + n] = fmaxf(acc, 0.f);
    }
}

// w1enc = enc @ w1^T + b1 as a WMMA GEMM: 4096 M-tiles of 16 rows.
__global__ __launch_bounds__(256) void w1enc_kernel(const float* enc,
        const float* W, const float* B, float* o) {
    __shared__ __align__(16) float arow[16][HD];
    const int tid = threadIdx.x, wvi = tid >> 5, lane = tid & 31;
    const size_t r0 = (size_t)blockIdx.x * 16;       // 65536 rows / 16
    lds_copy_issue((float*)arow, enc + r0 * HD, 16 * HD);
    lds_copy_join();
    __syncthreads();
    for (int tI = wvi; tI < 16; tI += 8) {
        int n0 = tI * 16;
        v8f acc; float bv = B[n0 + (tid & 15)];
#pragma unroll
        for (int r = 0; r < 8; ++r) acc[r] = bv;
        acc = wmma_acc((const float*)arow, HD, W, HD, n0, 0, HD, acc);
        const int nl = lane & 15, r8 = (lane >> 4) * 8;
#pragma unroll
        for (int r = 0; r < 8; ++r) o[(r0 + r8 + r) * HD + n0 + nl] = acc[r];
    }
}

__global__ void init_kernel(unsigned* bar) { bar[threadIdx.x] = 0u; }

// ---------------------------------------------------------------------------
extern "C" void kernel_launch(void* const* d_in, const int* in_sizes, int n_in,
                              void* d_out, int out_size, void* d_ws, size_t ws_size,
                              hipStream_t stream) {
    (void)in_sizes; (void)n_in; (void)out_size; (void)ws_size;
    const float* enc    = (const float*)d_in[0];
    const float* dec    = (const float*)d_in[1];
    const float* pre_w1 = (const float*)d_in[2];
    const float* pre_b1 = (const float*)d_in[3];
    const float* pre_w2 = (const float*)d_in[4];
    const float* pre_b2 = (const float*)d_in[5];
    const float* w1     = (const float*)d_in[6];
    const float* b1     = (const float*)d_in[7];

    Params P;
    P.enc = enc; P.xsrc_unused = nullptr;
    P.w2 = (const float*)d_in[8];   P.b2 = (const float*)d_in[9];
    P.v_w = (const float*)d_in[10]; P.v_b = (const float*)d_in[11];
    P.proj_w = (const float*)d_in[12]; P.proj_b = (const float*)d_in[13];
    P.out_w = (const float*)d_in[14];  P.out_b = (const float*)d_in[15];
    P.att_wih = (const float*)d_in[16]; P.att_whh = (const float*)d_in[17];
    P.att_bih = (const float*)d_in[18]; P.att_bhh = (const float*)d_in[19];
    P.g1_wih = (const float*)d_in[20]; P.g1_whh = (const float*)d_in[21];
    P.g1_bih = (const float*)d_in[22]; P.g1_bhh = (const float*)d_in[23];
    P.g2_wih = (const float*)d_in[24]; P.g2_whh = (const float*)d_in[25];
    P.g2_bih = (const float*)d_in[26]; P.g2_bhh = (const float*)d_in[27];

    float* ws = (float*)d_ws;
    P.xs    = ws;                                     // 3,276,800 floats
    P.w1enc = ws + 3276800;                           // 16,777,216 floats
    P.gws   = P.w1enc + 16777216;                     // NGROUP * GSZ_
    P.bar   = (unsigned*)(P.gws + (size_t)NGROUP * GSZ_);
    P.out   = (float*)d_out;

    init_kernel<<<1, 256, 0, stream>>>(P.bar);
    prenet_kernel<<<3200, 256, 0, stream>>>(dec, pre_w1, pre_b1, pre_w2, pre_b2, P.xs);
    w1enc_kernel<<<4096, 256, 0, stream>>>(enc, w1, b1, P.w1enc);
    scan_kernel<<<128, 256, 0, stream>>>(P);
}